// DirectionalAttention_86165633892831
// MI455X (gfx1250) — compile-verified
//
#include <hip/hip_runtime.h>
#include <math.h>

// Problem constants (match reference)
#define B_   8
#define S_   1024
#define D_   1024
#define H_   16
#define DK_  64
#define SCALE_  0.125f     // 1/sqrt(DK)
#define LN_EPS_ 1e-5f
#define SCP  1034          // LDS row stride (floats): 1034%64=10 -> 16 distinct banks, even -> 8B aligned

typedef __attribute__((ext_vector_type(2))) float v2f;
typedef __attribute__((ext_vector_type(8))) float v8f;
typedef __attribute__((ext_vector_type(4))) unsigned int u32x4;
typedef __attribute__((ext_vector_type(8))) int i32x8;
typedef __attribute__((ext_vector_type(4))) int i32x4;

// Native fp32 WMMA: D(16x16,f32) = A(16x4,f32) * B(4x16,f32) + C
__device__ __forceinline__ v8f wmma4(v2f a, v2f b, v8f c) {
  return __builtin_amdgcn_wmma_f32_16x16x4_f32(false, a, false, b, (short)0, c,
                                               false, false);
}

// ---------------------------------------------------------------------------
// Tensor Data Mover: build a 2-D tile descriptor (D#) and issue a DMA
// load global->LDS.  Bit packing per CDNA5 ISA 8.3/8.4.
//   pad_interval code 5 = every 64 dwords, pad_amount code 1 = 2 dwords
//   -> LDS row stride = tile_d0 + 2 floats.
// This toolchain exposes the 6-arg builtin:
//   (uint32x4 g0, int32x8 g1, int32x4 g2, int32x4 g3, int32x8 pad, i32 cpol)
// ---------------------------------------------------------------------------
__device__ __forceinline__ void tdm_load_2d(
    unsigned long long gaddr, unsigned int lds_addr,
    unsigned int tensor_d0, unsigned int tensor_d1,
    unsigned long long d0_stride,
    unsigned int tile_d0, unsigned int tile_d1,
    unsigned int pad_interval_code, unsigned int pad_amount_code) {
  u32x4 g0;
  g0[0] = 1u;                                            // count=1, user desc
  g0[1] = lds_addr;                                      // lds_addr [63:32]
  g0[2] = (unsigned int)(gaddr & 0xffffffffull);         // global_addr[31:0]
  g0[3] = (unsigned int)((gaddr >> 32) & 0x01ffffffull)  // global_addr[56:32]
        | (2u << 30);                                    // type = 2 ("image")
  i32x8 g1;
  g1[0] = (int)((2u << 16)                               // data_size = 4B
              | (1u << 20)                               // pad_enable
              | (pad_interval_code << 22)
              | (pad_amount_code << 25));
  g1[1] = (int)((tensor_d0 & 0xffffu) << 16);            // tensor_dim0[15:0]
  g1[2] = (int)(((tensor_d0 >> 16) & 0xffffu)            // tensor_dim0[31:16]
              | ((tensor_d1 & 0xffffu) << 16));          // tensor_dim1[15:0]
  g1[3] = (int)(((tensor_d1 >> 16) & 0xffffu)            // tensor_dim1[31:16]
              | ((tile_d0 & 0xffffu) << 16));            // tile_dim0
  g1[4] = (int)(tile_d1 & 0xffffu);                      // tile_dim1; tile_dim2=0
  g1[5] = (int)(unsigned int)(d0_stride & 0xffffffffull);// dim0_stride[31:0]
  g1[6] = (int)((d0_stride >> 32) & 0xffffull);          // dim0_stride[47:32]
  g1[7] = 0;                                             // dim1_stride unused (2D)
  i32x4 z4 = {0, 0, 0, 0};
  i32x8 z8 = {0, 0, 0, 0, 0, 0, 0, 0};
  __builtin_amdgcn_tensor_load_to_lds(g0, g1, z4, z4, z8, 0);
}

// ---------------------------------------------------------------------------
// Kernel 1: X[8192,1024] @ W^T + bias -> out in [B,H,S,DK] layout.
// Block (8 waves) computes a 64x256 tile; each wave a 32x64 sub-tile
// (2x4 fragment grid).  X/W k-chunks staged in LDS by the TDM.
// ---------------------------------------------------------------------------
#define KC     64                 // k-chunk (floats)
#define LDSTR  66                 // KC + 2 pad dwords (TDM pad)
#define XROWS  64
#define WROWS  256

__global__ __launch_bounds__(256) void proj_kernel(
    const float* __restrict__ X, const float* __restrict__ W,
    const float* __restrict__ bias, float* __restrict__ out) {
  extern __shared__ float lds[];
  float* Xs = lds;                         // [64][66]
  float* Ws = lds + XROWS * LDSTR;         // [256][66]
  const int tid = threadIdx.x;
  const int w = tid >> 5, lane = tid & 31;
  const int lm = lane & 15;
  const int kh = (lane >> 4) << 1;
  const int mtb = blockIdx.x >> 2;         // 64-row block (0..127)
  const int ntb = blockIdx.x & 3;          // 256-col block (0..3)
  const int mh = w >> 2;                   // 32-row half within block
  const int nq = w & 3;                    // 64-col quarter within block

  const unsigned int xs_off = (unsigned int)(size_t)Xs;  // LDS byte offsets
  const unsigned int ws_off = (unsigned int)(size_t)Ws;

  v8f acc[2][4] = {};

  for (int k0 = 0; k0 < D_; k0 += KC) {
    if (w == 0) {
      // X tile: 64 rows x KC cols starting (mtb*64, k0)
      tdm_load_2d((unsigned long long)(size_t)(X + (size_t)mtb * 64 * D_ + k0),
                  xs_off, D_, (unsigned)(B_ * S_), D_, KC, XROWS, 5, 1);
      // W tile: 256 rows x KC cols starting (ntb*256, k0)
      tdm_load_2d((unsigned long long)(size_t)(W + (size_t)ntb * 256 * D_ + k0),
                  ws_off, D_, D_, D_, KC, WROWS, 5, 1);
      __builtin_amdgcn_s_wait_tensorcnt(0);
    }
    __syncthreads();                       // tiles visible to all waves

    const float* xa0 = Xs + (mh * 32 + lm) * LDSTR;
    const float* xa1 = xa0 + 16 * LDSTR;
    const float* wb  = Ws + (nq * 64 + lm) * LDSTR;
    #pragma unroll 4
    for (int kc = 0; kc < KC; kc += 4) {
      v2f a0 = *(const v2f*)(xa0 + kc + kh);
      v2f a1 = *(const v2f*)(xa1 + kc + kh);
      v2f b0 = *(const v2f*)(wb + 0 * 16 * LDSTR + kc + kh);
      v2f b1 = *(const v2f*)(wb + 1 * 16 * LDSTR + kc + kh);
      v2f b2 = *(const v2f*)(wb + 2 * 16 * LDSTR + kc + kh);
      v2f b3 = *(const v2f*)(wb + 3 * 16 * LDSTR + kc + kh);
      acc[0][0] = wmma4(a0, b0, acc[0][0]);
      acc[0][1] = wmma4(a0, b1, acc[0][1]);
      acc[0][2] = wmma4(a0, b2, acc[0][2]);
      acc[0][3] = wmma4(a0, b3, acc[0][3]);
      acc[1][0] = wmma4(a1, b0, acc[1][0]);
      acc[1][1] = wmma4(a1, b1, acc[1][1]);
      acc[1][2] = wmma4(a1, b2, acc[1][2]);
      acc[1][3] = wmma4(a1, b3, acc[1][3]);
    }
    __syncthreads();                       // reads done before next DMA
  }

  // Epilogue: bias + store into [B,H,S,DK]
  #pragma unroll
  for (int j = 0; j < 4; ++j) {
    const int col = ntb * 256 + nq * 64 + j * 16 + lm;
    const float bn = bias[col];
    const int h = col >> 6, dk = col & 63;
    #pragma unroll
    for (int i = 0; i < 2; ++i) {
      #pragma unroll
      for (int e = 0; e < 8; ++e) {
        const int rowM = e + ((lane >> 4) << 3);
        const int r = mtb * 64 + mh * 32 + i * 16 + rowM;
        const int b = r >> 10, s = r & 1023;
        out[(((size_t)b * H_ + h) * S_ + s) * DK_ + dk] = acc[i][j][e] + bn;
      }
    }
  }
}

// ---------------------------------------------------------------------------
// Kernel 2: per (b,h, q-tile of 16 rows): scores -> LDS, softmax in LDS,
// attn written once, O = P @ V via WMMA out of LDS.
// ---------------------------------------------------------------------------
__global__ __launch_bounds__(256) void attn_kernel(
    const float* __restrict__ Q, const float* __restrict__ K,
    const float* __restrict__ V, float* __restrict__ attn,
    float* __restrict__ O) {
  extern __shared__ float lds[];
  float* sc    = lds;               // [16][SCP]
  float* opart = lds + 16 * SCP;    // [8][16][16]
  __shared__ float red[16][17];

  const int bh = blockIdx.x >> 6;
  const int qt = blockIdx.x & 63;
  const int b  = bh >> 4, h = bh & 15;
  const int tid = threadIdx.x;
  const int w = tid >> 5, lane = tid & 31;
  const int lm = lane & 15;
  const int kh = (lane >> 4) << 1;

  const float* Qp = Q + ((size_t)bh * S_ + qt * 16) * DK_;
  const float* Kp = K + (size_t)bh * S_ * DK_;
  const float* Vp = V + (size_t)bh * S_ * DK_;

  // Q tile (16x64) is reused across all 64 K tiles: preload fragments.
  v2f qf[16];
  #pragma unroll
  for (int i = 0; i < 16; ++i)
    qf[i] = *(const v2f*)(Qp + lm * DK_ + i * 4 + kh);

  // ---- Phase 1: 64 score tiles, 8 per wave ----
  for (int t = 0; t < 8; ++t) {
    const int kt = w * 8 + t;
    const float* brow = Kp + (size_t)(kt * 16 + lm) * DK_;
    v8f acc = {};
    #pragma unroll
    for (int i = 0; i < 16; ++i)
      acc = wmma4(qf[i], *(const v2f*)(brow + i * 4 + kh), acc);
    #pragma unroll
    for (int i = 0; i < 8; ++i) {
      const int rowM = i + ((lane >> 4) << 3);
      const int qg = qt * 16 + rowM;
      const int kg = kt * 16 + lm;
      float v = acc[i] * SCALE_;
      if (kg > qg) v = -1e9f;       // causal: mask==0 -> -1e9
      sc[rowM * SCP + kg] = v;
    }
  }
  __syncthreads();

  // ---- Phase 2: row softmax (16 threads per row) ----
  const int r = tid >> 4, sub = tid & 15;
  float mx = -3.0e38f;
  for (int c = sub; c < S_; c += 16) mx = fmaxf(mx, sc[r * SCP + c]);
  red[r][sub] = mx;
  __syncthreads();
  if (sub == 0) {
    float m = red[r][0];
    for (int i = 1; i < 16; ++i) m = fmaxf(m, red[r][i]);
    red[r][16] = m;
  }
  __syncthreads();
  const float rowmax = red[r][16];
  float sum = 0.f;
  for (int c = sub; c < S_; c += 16) {
    float e = __expf(sc[r * SCP + c] - rowmax);
    sc[r * SCP + c] = e;
    sum += e;
  }
  red[r][sub] = sum;
  __syncthreads();
  if (sub == 0) {
    float s2 = 0.f;
    for (int i = 0; i < 16; ++i) s2 += red[r][i];
    red[r][16] = 1.0f / s2;
  }
  __syncthreads();
  const float inv = red[r][16];
  float* attnrow = attn + ((size_t)bh * S_ + qt * 16 + r) * S_;
  for (int c = sub; c < S_; c += 16) {
    float p = sc[r * SCP + c] * inv;
    sc[r * SCP + c] = p;
    attnrow[c] = p;
  }
  __syncthreads();

  // ---- Phase 3: O(16x64) = P(16x1024) @ V(1024x64) ----
  {
    const int ntile = w & 3;
    const int khalf = w >> 2;
    const int kbeg = khalf * 512, kend = kbeg + 512;
    v8f acc = {};
    for (int kk = kbeg; kk < kend; kk += 4) {
      v2f a = *(const v2f*)(sc + lm * SCP + kk + kh);
      v2f bb;
      bb.x = Vp[(size_t)(kk + kh)     * DK_ + ntile * 16 + lm];
      bb.y = Vp[(size_t)(kk + kh + 1) * DK_ + ntile * 16 + lm];
      acc = wmma4(a, bb, acc);
    }
    #pragma unroll
    for (int i = 0; i < 8; ++i) {
      const int rowM = i + ((lane >> 4) << 3);
      opart[w * 256 + rowM * 16 + lm] = acc[i];
    }
  }
  __syncthreads();
  for (int e = tid; e < 16 * 64; e += 256) {
    const int rr = e >> 6, c = e & 63;
    const int nt2 = c >> 4, cc = c & 15;
    float v = opart[nt2 * 256 + rr * 16 + cc] +
              opart[(4 + nt2) * 256 + rr * 16 + cc];
    O[((size_t)b * S_ + qt * 16 + rr) * D_ + h * DK_ + c] = v;
  }
}

// ---------------------------------------------------------------------------
// Kernel 3: out = O @ wo^T + bo; x = out + query; y = LayerNorm(x).
// 1x4 register blocking: one A fragment feeds 4 WMMAs.
// ---------------------------------------------------------------------------
__global__ __launch_bounds__(256) void oproj_ln_kernel(
    const float* __restrict__ O, const float* __restrict__ Wo,
    const float* __restrict__ bo, const float* __restrict__ query,
    const float* __restrict__ gamma, const float* __restrict__ beta,
    float* __restrict__ Y) {
  extern __shared__ float xln[];    // [16][SCP]
  __shared__ float red[16][17];
  const int rt  = blockIdx.x;
  const int tid = threadIdx.x;
  const int w = tid >> 5, lane = tid & 31;
  const int lm = lane & 15, kh = (lane >> 4) << 1;

  const float* arow = O + (size_t)(rt * 16 + lm) * D_;
  for (int g = 0; g < 2; ++g) {
    const int nt0 = w * 8 + g * 4;
    const float* br0 = Wo + (size_t)(nt0 * 16 + lm) * D_;
    const float* br1 = br0 + (size_t)16 * D_;
    const float* br2 = br1 + (size_t)16 * D_;
    const float* br3 = br2 + (size_t)16 * D_;
    v8f acc[4] = {};
    #pragma unroll 4
    for (int k0 = 0; k0 < D_; k0 += 4) {
      v2f a = *(const v2f*)(arow + k0 + kh);
      acc[0] = wmma4(a, *(const v2f*)(br0 + k0 + kh), acc[0]);
      acc[1] = wmma4(a, *(const v2f*)(br1 + k0 + kh), acc[1]);
      acc[2] = wmma4(a, *(const v2f*)(br2 + k0 + kh), acc[2]);
      acc[3] = wmma4(a, *(const v2f*)(br3 + k0 + kh), acc[3]);
    }
    #pragma unroll
    for (int j = 0; j < 4; ++j) {
      const int n = (nt0 + j) * 16 + lm;
      const float bn = bo[n];
      #pragma unroll
      for (int e = 0; e < 8; ++e) {
        const int rowM = e + ((lane >> 4) << 3);
        xln[rowM * SCP + n] =
            acc[j][e] + bn + query[(size_t)(rt * 16 + rowM) * D_ + n];
      }
    }
  }
  __syncthreads();

  const int r = tid >> 4, sub = tid & 15;
  float s = 0.f;
  for (int c = sub; c < D_; c += 16) s += xln[r * SCP + c];
  red[r][sub] = s;
  __syncthreads();
  if (sub == 0) {
    float t2 = 0.f;
    for (int i = 0; i < 16; ++i) t2 += red[r][i];
    red[r][16] = t2 * (1.0f / D_);
  }
  __syncthreads();
  const float mu = red[r][16];
  float vs = 0.f;
  for (int c = sub; c < D_; c += 16) {
    float d = xln[r * SCP + c] - mu;
    vs += d * d;
  }
  red[r][sub] = vs;
  __syncthreads();
  if (sub == 0) {
    float t2 = 0.f;
    for (int i = 0; i < 16; ++i) t2 += red[r][i];
    red[r][16] = rsqrtf(t2 * (1.0f / D_) + LN_EPS_);
  }
  __syncthreads();
  const float rstd = red[r][16];
  float* yrow = Y + (size_t)(rt * 16 + r) * D_;
  for (int c = sub; c < D_; c += 16)
    yrow[c] = (xln[r * SCP + c] - mu) * rstd * gamma[c] + beta[c];
}

// ---------------------------------------------------------------------------
extern "C" void kernel_launch(void* const* d_in, const int* in_sizes, int n_in,
                              void* d_out, int out_size, void* d_ws, size_t ws_size,
                              hipStream_t stream) {
  const float* query = (const float*)d_in[0];
  const float* key   = (const float*)d_in[1];
  const float* value = (const float*)d_in[2];
  // d_in[3] = causal_mask (tril) — applied analytically as (kcol > qrow)
  const float* wq = (const float*)d_in[4];
  const float* bq = (const float*)d_in[5];
  const float* wk = (const float*)d_in[6];
  const float* bk = (const float*)d_in[7];
  const float* wv = (const float*)d_in[8];
  const float* bv = (const float*)d_in[9];
  const float* wo = (const float*)d_in[10];
  const float* bo = (const float*)d_in[11];
  const float* gamma = (const float*)d_in[12];
  const float* beta  = (const float*)d_in[13];

  const size_t MAT = (size_t)B_ * S_ * D_;
  float* Q = (float*)d_ws;
  float* K = Q + MAT;
  float* V = K + MAT;
  float* O = V + MAT;

  float* y    = (float*)d_out;               // [B,S,D]
  float* attn = y + MAT;                     // [B,H,S,S]

  dim3 blk(256);
  // 128 row-blocks x 4 col-blocks = 512 blocks per matrix
  const size_t lds1 = (size_t)(XROWS * LDSTR + WROWS * LDSTR) * sizeof(float);
  proj_kernel<<<512, blk, lds1, stream>>>(query, wq, bq, Q);
  proj_kernel<<<512, blk, lds1, stream>>>(key,   wk, bk, K);
  proj_kernel<<<512, blk, lds1, stream>>>(value, wv, bv, V);

  const size_t lds2 = (size_t)(16 * SCP + 8 * 16 * 16) * sizeof(float);
  attn_kernel<<<B_ * H_ * (S_ / 16), blk, lds2, stream>>>(Q, K, V, attn, O);

  const size_t lds3 = (size_t)(16 * SCP) * sizeof(float);
  oproj_ln_kernel<<<(B_ * S_) / 16, blk, lds3, stream>>>(O, wo, bo, query,
                                                         gamma, beta, y);
}